// GNN_75093208203518
// MI455X (gfx1250) — compile-verified
//
#include <hip/hip_runtime.h>
#include <hip/hip_bf16.h>

typedef float v2f __attribute__((ext_vector_type(2)));
typedef float v8f __attribute__((ext_vector_type(8)));

#define N_NODES 100000
#define D_FEAT  64

// ---------------- degree / normalization ----------------

__global__ void init_deg_kernel(float* deg, int n) {
    int i = blockIdx.x * blockDim.x + threadIdx.x;
    if (i < n) deg[i] = 1.0f;  // self loop
}

__global__ void count_deg_kernel(const long long* __restrict__ dst, float* deg, int e) {
    int i = blockIdx.x * blockDim.x + threadIdx.x;
    if (i < e) atomicAdd(&deg[(int)dst[i]], 1.0f);
}

__global__ void rsqrt_kernel(float* dinv, int n) {
    int i = blockIdx.x * blockDim.x + threadIdx.x;
    if (i < n) dinv[i] = rsqrtf(dinv[i]);
}

// ---------------- GEMM: H = X * W  (fp32 WMMA 16x16x4) ----------------
// One wave computes a 16-row x 64-col output stripe.
// A (16x4 f32): lanes 0-15 hold M=lane, K=k0/k0+1; lanes 16-31 hold M=lane-16, K=k0+2/k0+3.
// B (4x16 f32): lanes 0-15 hold K=k0/k0+1 rows, lanes 16-31 hold K=k0+2/k0+3 rows (N=lane&15).
// C/D (16x16 f32): VGPR v: lanes 0-15 -> M=v, lanes 16-31 -> M=v+8, N=lane&15.

__global__ void __launch_bounds__(256)
gemm_wmma_kernel(const float* __restrict__ X, const float* __restrict__ W,
                 float* __restrict__ H, int nRows) {
    int wave = (int)((blockIdx.x * blockDim.x + threadIdx.x) >> 5);
    int lane = threadIdx.x & 31;
    int rowTile = wave;
    if (rowTile * 16 >= nRows) return;   // whole-wave uniform: EXEC stays all-1s

    const int m     = rowTile * 16 + (lane & 15);
    const int khalf = (lane >> 4) * 2;   // 0 or 2
    const int col   = lane & 15;

    v8f acc0 = {}, acc1 = {}, acc2 = {}, acc3 = {};

    for (int k0 = 0; k0 < D_FEAT; k0 += 4) {
        const int ka = k0 + khalf;
        v2f a;
        a.x = X[(long long)m * D_FEAT + ka];
        a.y = X[(long long)m * D_FEAT + ka + 1];

        const float* Wr0 = W + ka * D_FEAT;
        const float* Wr1 = W + (ka + 1) * D_FEAT;
        v2f b0, b1, b2, b3;
        b0.x = Wr0[col +  0]; b0.y = Wr1[col +  0];
        b1.x = Wr0[col + 16]; b1.y = Wr1[col + 16];
        b2.x = Wr0[col + 32]; b2.y = Wr1[col + 32];
        b3.x = Wr0[col + 48]; b3.y = Wr1[col + 48];

        acc0 = __builtin_amdgcn_wmma_f32_16x16x4_f32(false, a, false, b0, (short)0, acc0, false, false);
        acc1 = __builtin_amdgcn_wmma_f32_16x16x4_f32(false, a, false, b1, (short)0, acc1, false, false);
        acc2 = __builtin_amdgcn_wmma_f32_16x16x4_f32(false, a, false, b2, (short)0, acc2, false, false);
        acc3 = __builtin_amdgcn_wmma_f32_16x16x4_f32(false, a, false, b3, (short)0, acc3, false, false);
    }

    const int mBase = rowTile * 16 + (lane >> 4) * 8;
    const int n     = lane & 15;
#pragma unroll
    for (int v = 0; v < 8; ++v) {
        long long r = (long long)(mBase + v) * D_FEAT;
        H[r + n +  0] = acc0[v];
        H[r + n + 16] = acc1[v];
        H[r + n + 32] = acc2[v];
        H[r + n + 48] = acc3[v];
    }
}

// ---------------- aggregation ----------------

// agg[i,:] = dinv[i]^2 * h[i,:]   (self-loop term; also initializes agg)
__global__ void self_loop_kernel(const float* __restrict__ h, const float* __restrict__ dinv,
                                 float* __restrict__ agg, int total) {
    int t = blockIdx.x * blockDim.x + threadIdx.x;
    if (t < total) {
        int i = t >> 6;
        float dv = dinv[i];
        agg[t] = dv * dv * h[t];
    }
}

// agg[dst,:] += dinv[src]*dinv[dst] * h[src,:]   -- 64 lanes (2 waves) per edge, coalesced atomics
__global__ void scatter_kernel(const float* __restrict__ h,
                               const long long* __restrict__ src,
                               const long long* __restrict__ dst,
                               const float* __restrict__ dinv,
                               float* __restrict__ agg, int e) {
    long long t = (long long)blockIdx.x * blockDim.x + threadIdx.x;
    long long total = (long long)e * D_FEAT;
    if (t < total) {
        long long ed = t >> 6;
        int d = (int)(t & 63);
        long long s  = src[ed];
        long long dd = dst[ed];
        float norm = dinv[(int)s] * dinv[(int)dd];
        atomicAdd(&agg[dd * D_FEAT + d], h[s * D_FEAT + d] * norm);
    }
}

// out = relu(agg + b)
__global__ void bias_relu_kernel(const float* __restrict__ agg, const float* __restrict__ b,
                                 float* __restrict__ out, int total) {
    int t = blockIdx.x * blockDim.x + threadIdx.x;
    if (t < total) {
        float v = agg[t] + b[t & 63];
        out[t] = v > 0.0f ? v : 0.0f;
    }
}

extern "C" void kernel_launch(void* const* d_in, const int* in_sizes, int n_in,
                              void* d_out, int out_size, void* d_ws, size_t ws_size,
                              hipStream_t stream) {
    const float*     x    = (const float*)d_in[0];
    const long long* eidx = (const long long*)d_in[1];   // int64 per reference
    const float*     W1   = (const float*)d_in[2];
    const float*     b1   = (const float*)d_in[3];
    const float*     W2   = (const float*)d_in[4];
    const float*     b2   = (const float*)d_in[5];
    float*           out  = (float*)d_out;

    const int N = in_sizes[0] / D_FEAT;      // 100000
    const int E = in_sizes[1] / 2;           // 1600000
    const long long* src = eidx;
    const long long* dst = eidx + E;

    // workspace layout (floats)
    float* ws   = (float*)d_ws;
    float* dinv = ws;                                  // N
    float* h    = ws + 102400;                         // N*64
    float* agg  = h  + (size_t)N * D_FEAT;             // N*64
    float* h2   = agg + (size_t)N * D_FEAT;            // N*64

    const int nd  = N * D_FEAT;                        // 6.4M
    const long long ewk = (long long)E * D_FEAT;       // 102.4M

    const int TB = 256;
    dim3 blkN((N + TB - 1) / TB);
    dim3 blkE((E + TB - 1) / TB);
    dim3 blkND((nd + TB - 1) / TB);
    dim3 blkEW((unsigned)((ewk + TB - 1) / TB));
    // GEMM: one wave per 16-row tile, 8 waves per block
    int rowTiles = (N + 15) / 16;                      // 6250
    dim3 blkG((rowTiles + 7) / 8);

    // normalization (once)
    init_deg_kernel  <<<blkN,  TB, 0, stream>>>(dinv, N);
    count_deg_kernel <<<blkE,  TB, 0, stream>>>(dst, dinv, E);
    rsqrt_kernel     <<<blkN,  TB, 0, stream>>>(dinv, N);

    // ---- layer 1 ----
    gemm_wmma_kernel <<<blkG,  TB, 0, stream>>>(x, W1, h, N);
    self_loop_kernel <<<blkND, TB, 0, stream>>>(h, dinv, agg, nd);
    scatter_kernel   <<<blkEW, TB, 0, stream>>>(h, src, dst, dinv, agg, E);
    bias_relu_kernel <<<blkND, TB, 0, stream>>>(agg, b1, h2, nd);

    // ---- layer 2 ----
    gemm_wmma_kernel <<<blkG,  TB, 0, stream>>>(h2, W2, h, N);
    self_loop_kernel <<<blkND, TB, 0, stream>>>(h, dinv, agg, nd);
    scatter_kernel   <<<blkEW, TB, 0, stream>>>(h, src, dst, dinv, agg, E);
    bias_relu_kernel <<<blkND, TB, 0, stream>>>(agg, b2, out, nd);
}